// CombinedRankingLoss_40355512713933
// MI455X (gfx1250) — compile-verified
//
#include <hip/hip_runtime.h>

typedef __attribute__((ext_vector_type(2))) float v2f;
typedef __attribute__((ext_vector_type(8))) float v8f;

#define ROW_N 2048
#define NBLK  256
#define CHUNK (ROW_N / NBLK) /* 8 */
#define NWAVE (NBLK / 32)    /* 8 wave32 waves */

#define LOG2E 1.4426950408889634f
#define LN2   0.6931471805599453f

__device__ __forceinline__ float block_sum(float v, float* red, int t) {
  red[t] = v; __syncthreads();
  for (int off = NBLK / 2; off > 0; off >>= 1) {
    if (t < off) red[t] += red[t + off];
    __syncthreads();
  }
  float r = red[0];
  __syncthreads();
  return r;
}

__device__ __forceinline__ float block_max(float v, float* red, int t) {
  red[t] = v; __syncthreads();
  for (int off = NBLK / 2; off > 0; off >>= 1) {
    if (t < off) red[t] = fmaxf(red[t], red[t + off]);
    __syncthreads();
  }
  float r = red[0];
  __syncthreads();
  return r;
}

__global__ __launch_bounds__(NBLK)
void crl_row_kernel(const float* __restrict__ scores,
                    const int*   __restrict__ rel,
                    float*       __restrict__ row_out) {
  __shared__ float ss[ROW_N];      // stable-compacted scores: positives first, then negatives
  __shared__ float ex[ROW_N];      // exp(ss - M) for suffix sums
  __shared__ float red[NBLK];      // reduction scratch
  __shared__ int   iscan[NBLK];
  __shared__ float fscan[NBLK];

  const int b = blockIdx.x;
  const int t = threadIdx.x;
  const float* srow = scores + b * ROW_N;
  const int*   yrow = rel    + b * ROW_N;
  const int c0 = t * CHUNK;

  // ---- Phase A: load chunk, focal partial, max partial, pos counts ----
  float sv[CHUNK];
  int   fl[CHUNK];
  int   cnt  = 0;
  float fsum = 0.0f;
  float m    = -3.0e38f;
#pragma unroll
  for (int k = 0; k < CHUNK; ++k) {
    float s = srow[c0 + k];
    int   y = yrow[c0 + k];
    sv[k] = s;
    fl[k] = (y == 1);
    m = fmaxf(m, s);
    if (fl[k]) {
      ++cnt;
      // p = sigmoid(s) via raw 2^x and rcp (args always in safe range)
      float p = __builtin_amdgcn_rcpf(1.0f + __builtin_amdgcn_exp2f(-LOG2E * s));
      float q = 1.0f - p;
      // log(p + 1e-8) = ln2 * log2(p + 1e-8); arg >= 1e-8 (normal), no fixups needed
      fsum += -(q * q) * (LN2 * __builtin_amdgcn_logf(p + 1e-8f));
    }
  }

  // inclusive Hillis-Steele scan of per-chunk positive counts (8 steps)
  iscan[t] = cnt; __syncthreads();
#pragma unroll
  for (int off = 1; off < NBLK; off <<= 1) {
    int v = (t >= off) ? iscan[t - off] : 0;
    __syncthreads();
    iscan[t] += v;
    __syncthreads();
  }
  const int npos = iscan[NBLK - 1];
  const int nneg = ROW_N - npos;
  const int posBefore = iscan[t] - cnt;   // exclusive

  // stable scatter: positives to [0,npos), negatives to [npos,N), original order kept
  int wp = posBefore;
  int wn = npos + (c0 - posBefore);
#pragma unroll
  for (int k = 0; k < CHUNK; ++k) {
    if (fl[k]) ss[wp++] = sv[k];
    else       ss[wn++] = sv[k];
  }

  const float M         = block_max(m, red, t);  // barrier publishes ss[]
  const float foc_total = block_sum(fsum, red, t);

  // ---- Phase B: ListMLE suffix logsumexp over compacted ss ----
  const float Ml2 = M * LOG2E;
  float csum = 0.0f;
#pragma unroll
  for (int k = 0; k < CHUNK; ++k) {
    // exp(ss - M) = 2^(ss*log2e - M*log2e); arg <= 0, result in (0,1]: raw exp2 is safe
    float e = __builtin_amdgcn_exp2f(fmaf(ss[c0 + k], LOG2E, -Ml2));
    ex[c0 + k] = e;
    csum += e;
  }
  // reverse inclusive scan of chunk sums (suffix sums), 8 steps
  fscan[t] = csum; __syncthreads();
#pragma unroll
  for (int off = 1; off < NBLK; off <<= 1) {
    float v = (t + off < NBLK) ? fscan[t + off] : 0.0f;
    __syncthreads();
    fscan[t] += v;
    __syncthreads();
  }
  float run  = fscan[t] - csum;  // sum of all elements strictly after this chunk
  float part = 0.0f;
#pragma unroll
  for (int k = CHUNK - 1; k >= 0; --k) {
    int idx = c0 + k;
    run += ex[idx];  // run = S_idx = sum_{j>=idx} exp(ss_j - M), in [exp(min-M), 2048]
    if (idx <= ROW_N - 2) part += (M + LN2 * __builtin_amdgcn_logf(run)) - ss[idx];
  }
  const float lml_total = block_sum(part, red, t);

  // ---- Phase C: LambdaRank pairwise softplus via WMMA rank-1 diff tiles ----
  // D[i][j] = Ng[j] - P[i] on the matrix pipe:
  //   A(16x4): col K=0 = -P[i0..i0+15];  B(4x16): row K=0 = ones;  C: Ng[j] per column
  // Padding P with +1000 / Ng with -1000 => softplus contribution exactly 0.
  // Per-tile sum uses log batching + abs-sum identity:
  //   sum softplus(x_v) = 0.5*(sum x_v + sum |x_v|)
  //                     + ln2 * log2( prod_v (1 + 2^(-log2e*|x_v|)) )
  // Each factor in [1,2] => product of 8 in [1,256]: exact f32 range, single log per tile.
  // Dual accumulators keep the FMA/add dependency chains at depth 4.
  const int lane = t & 31;
  const int wv   = t >> 5;
  const int nti  = (npos + 15) >> 4;
  const int ntj  = (nneg + 15) >> 4;
  const int ntiles = nti * ntj;

  float lsum = 0.0f;
  v2f bmat;
  bmat.x = (lane < 16) ? 1.0f : 0.0f;   // K=0 row of B is ones
  bmat.y = 0.0f;

  for (int tile = wv; tile < ntiles; tile += NWAVE) {
    int ti = tile / ntj;
    int tj = tile - ti * ntj;
    int i0 = ti << 4;
    int j0 = tj << 4;

    int   il = i0 + lane;
    float Pv = (lane < 16) ? ((il < npos) ? ss[il] : 1.0e3f) : 0.0f;
    v2f amat;
    amat.x = (lane < 16) ? -Pv : 0.0f;  // K=0 column of A = -P
    amat.y = 0.0f;

    int   jl = j0 + (lane & 15);
    float Nv = (jl < nneg) ? ss[npos + jl] : -1.0e3f;
    v8f c;
#pragma unroll
    for (int v = 0; v < 8; ++v) c[v] = Nv;  // C broadcasts Ng[j] down each column

    // EXEC all-ones here (selects are cndmask / reconverged)
    v8f d = __builtin_amdgcn_wmma_f32_16x16x4_f32(
        /*neg_a=*/false, amat, /*neg_b=*/false, bmat,
        /*c_mod=*/(short)0, c, /*reuse_a=*/false, /*reuse_b=*/false);

    float sx0 = 0.0f, sx1 = 0.0f;   // sum x     (|.| is a free src modifier)
    float sa0 = 0.0f, sa1 = 0.0f;   // sum |x|
    float pr0 = 1.0f, pr1 = 1.0f;   // partial products of (1 + 2^(-log2e*|x|))
#pragma unroll
    for (int v = 0; v < 8; v += 2) {
      float x0 = d[v], x1 = d[v + 1];
      float a0 = fabsf(x0), a1 = fabsf(x1);
      sx0 += x0;  sx1 += x1;
      sa0 += a0;  sa1 += a1;
      float t0 = __builtin_amdgcn_exp2f(-LOG2E * a0); // (0,1]
      float t1 = __builtin_amdgcn_exp2f(-LOG2E * a1);
      pr0 = fmaf(pr0, t0, pr0);                       // pr0 *= (1+t0)
      pr1 = fmaf(pr1, t1, pr1);                       // pr1 *= (1+t1)
    }
    lsum += 0.5f * ((sx0 + sx1) + (sa0 + sa1))
          + LN2 * __builtin_amdgcn_logf(pr0 * pr1);
  }
  const float lam_total = block_sum(lsum, red, t);

  if (t == 0) {
    float paircnt = (float)npos * (float)nneg;
    float lam = (paircnt > 0.0f) ? lam_total / paircnt : 0.0f;
    float lml = (npos > 0) ? lml_total / (float)(ROW_N - 1) : 0.0f;
    float foc = (npos > 0) ? foc_total / fmaxf((float)npos, 1.0f) : 0.0f;
    row_out[b] = 0.5f * lam + 0.3f * lml + 0.2f * foc;
  }
}

__global__ void crl_final_kernel(const float* __restrict__ row_loss,
                                 float* __restrict__ out, int nrows) {
  if (threadIdx.x == 0) {
    float s = 0.0f;
    for (int i = 0; i < nrows; ++i) s += row_loss[i];
    out[0] = s / (float)nrows;
  }
}

extern "C" void kernel_launch(void* const* d_in, const int* in_sizes, int n_in,
                              void* d_out, int out_size, void* d_ws, size_t ws_size,
                              hipStream_t stream) {
  const float* scores = (const float*)d_in[0];
  const int*   rel    = (const int*)d_in[1];
  float* out = (float*)d_out;
  float* ws  = (float*)d_ws;

  const int nrows = in_sizes[0] / ROW_N;  // B = 16

  crl_row_kernel<<<nrows, NBLK, 0, stream>>>(scores, rel, ws);
  crl_final_kernel<<<1, 32, 0, stream>>>(ws, out, nrows);
}